// DAMPLayer_22454089023534
// MI455X (gfx1250) — compile-verified
//
#include <hip/hip_runtime.h>
#include <hip/hip_bf16.h>
#include <math.h>

// ---------------------------------------------------------------------------
// DAMP graph-attention layer for gfx1250 (MI455X), fp32 WMMA 16x16x4 path.
// ---------------------------------------------------------------------------

typedef __attribute__((ext_vector_type(2))) float v2f;
typedef __attribute__((ext_vector_type(8))) float v8f;

#define N_NODES 10000
#define N_EDGES 640000
#define NF 64
#define EF 64
#define NH 128
#define EH 128
#define CS 128

__device__ __forceinline__ v8f wmma4(v2f a, v2f b, v8f c) {
  // D = A(16x4 f32) * B(4x16 f32) + C(16x16 f32)
  return __builtin_amdgcn_wmma_f32_16x16x4_f32(
      /*neg_a=*/false, a, /*neg_b=*/false, b,
      /*c_mod=*/(short)0, c, /*reuse_a=*/false, /*reuse_b=*/false);
}

__device__ __forceinline__ float lrelu(float x) { return x > 0.0f ? x : 0.01f * x; }

// order-preserving float <-> uint encoding for atomicMax on floats
__device__ __forceinline__ unsigned enc_f(float f) {
  unsigned u = __float_as_uint(f);
  return (u & 0x80000000u) ? ~u : (u | 0x80000000u);
}
__device__ __forceinline__ float dec_f(unsigned e) {
  return (e & 0x80000000u) ? __uint_as_float(e & 0x7FFFFFFFu)
                           : __uint_as_float(~e);
}

// ---------------------------------------------------------------------------
// init: zero C_v, den; lmax encoded-max buffer to 0 (== lowest encoding)
// ---------------------------------------------------------------------------
__global__ void init_k(float* __restrict__ Cv, float* __restrict__ den,
                       unsigned* __restrict__ lmax) {
  int g = blockIdx.x * blockDim.x + threadIdx.x;
  if (g < N_NODES * CS) Cv[g] = 0.0f;
  if (g < N_NODES) { den[g] = 0.0f; lmax[g] = 0u; }
}

// ---------------------------------------------------------------------------
// (1) h_v = LeakyReLU(node_feats @ W_node + b_node)   [10000 x 128]
// One workgroup = 8 waves, 16 nodes; wave w owns column tile w (16 cols).
// ---------------------------------------------------------------------------
__global__ __launch_bounds__(256) void node_embed_k(
    const float* __restrict__ nf, const float* __restrict__ Wn,
    const float* __restrict__ bn, float* __restrict__ hv) {
  int tid = threadIdx.x;
  int wave = tid >> 5, lane = tid & 31;
  int m = lane & 15, half = lane >> 4;
  int base = blockIdx.x * 16;
  int row = base + m;           // node row for A fragment
  int ncol = wave * 16 + m;     // output column (n = lane&15)
  v8f acc = {};
  for (int k = 0; k < NF; k += 4) {
    int ka = k + half * 2;
    v2f a, b;
    a.x = nf[row * NF + ka];
    a.y = nf[row * NF + ka + 1];
    b.x = Wn[ka * NH + ncol];
    b.y = Wn[(ka + 1) * NH + ncol];
    acc = wmma4(a, b, acc);
  }
  float bias = bn[ncol];
#pragma unroll
  for (int i = 0; i < 8; ++i) {
    int r = base + i + half * 8;
    hv[r * NH + ncol] = lrelu(acc[i] + bias);
  }
}

// ---------------------------------------------------------------------------
// (2)+(3) per 16-edge tile: h_e in LDS (WMMA over K=128 concat input),
// then attention logit = LeakyReLU([h_v[src]|h_e] . W_logit + b_logit)
// ---------------------------------------------------------------------------
__global__ __launch_bounds__(256) void edge_logit_k(
    const float* __restrict__ nf, const float* __restrict__ ef,
    const float* __restrict__ We, const float* __restrict__ be,
    const float* __restrict__ Wl, const float* __restrict__ bl,
    const float* __restrict__ hv, const int* __restrict__ src,
    float* __restrict__ logit) {
  __shared__ float ein[16][NF + EF];
  __shared__ float he[16][EH];
  __shared__ float lsum[16];
  int tid = threadIdx.x;
  int ebase = blockIdx.x * 16;
  int er = tid >> 4;   // edge row 0..15
  int seg = tid & 15;  // 16 segments of 8 cols
  int e = ebase + er;
  int s = src[e];
  if (seg < 8) {
    const float* p = nf + s * NF + seg * 8;
#pragma unroll
    for (int q = 0; q < 8; ++q) ein[er][seg * 8 + q] = p[q];
  } else {
    const float* p = ef + e * EF + (seg - 8) * 8;
#pragma unroll
    for (int q = 0; q < 8; ++q) ein[er][seg * 8 + q] = p[q];
  }
  if (tid < 16) lsum[tid] = 0.0f;
  __syncthreads();

  // h_e GEMM: 8 waves x (16x16 tile), K = 128
  int wave = tid >> 5, lane = tid & 31;
  int m = lane & 15, half = lane >> 4;
  int ncol = wave * 16 + m;
  v8f acc = {};
  for (int k = 0; k < NF + EF; k += 4) {
    int ka = k + half * 2;
    v2f a, b;
    a.x = ein[m][ka];
    a.y = ein[m][ka + 1];
    b.x = We[ka * EH + ncol];
    b.y = We[(ka + 1) * EH + ncol];
    acc = wmma4(a, b, acc);
  }
  float bias = be[ncol];
#pragma unroll
  for (int i = 0; i < 8; ++i) he[i + half * 8][ncol] = lrelu(acc[i] + bias);
  __syncthreads();

  // logit partial dot over 16 of 256 concat dims per thread
  float part = 0.0f;
  if (seg < 8) {
    const float* p = hv + s * NH + seg * 16;
    const float* w = Wl + seg * 16;
#pragma unroll
    for (int q = 0; q < 16; ++q) part += p[q] * w[q];
  } else {
    const float* w = Wl + seg * 16;  // dims 128..255
#pragma unroll
    for (int q = 0; q < 16; ++q) part += he[er][(seg - 8) * 16 + q] * w[q];
  }
  atomicAdd(&lsum[er], part);  // ds_add_f32
  __syncthreads();
  if (tid < 16) logit[ebase + tid] = lrelu(lsum[tid] + bl[0]);
}

// ---------------------------------------------------------------------------
// (4a) segment max via encoded atomicMax (lmax / C_v live in L2)
// ---------------------------------------------------------------------------
__global__ void seg_max_k(const float* __restrict__ logit,
                          const int* __restrict__ dst,
                          unsigned* __restrict__ lmax) {
  int g = blockIdx.x * blockDim.x + threadIdx.x;
  if (g < N_EDGES) atomicMax(&lmax[dst[g]], enc_f(logit[g]));
}

// (4b) ex = exp(logit - lmax[dst]); den[dst] += ex  (ex overwrites logit)
__global__ void seg_exp_k(float* __restrict__ logit,
                          const int* __restrict__ dst,
                          const unsigned* __restrict__ lmax,
                          float* __restrict__ den) {
  int g = blockIdx.x * blockDim.x + threadIdx.x;
  if (g < N_EDGES) {
    int d = dst[g];
    float ex = __expf(logit[g] - dec_f(lmax[d]));
    logit[g] = ex;
    atomicAdd(&den[d], ex);
  }
}

// ---------------------------------------------------------------------------
// (5) recompute h_e in LDS, msg GEMM, scale by alpha, scatter-add into C_v.
// Recompute beats spilling 656 MB of h_e/m to HBM (edge_feats may even be
// L2-warm: 164 MB vs 192 MB L2).
// ---------------------------------------------------------------------------
__global__ __launch_bounds__(256) void edge_scatter_k(
    const float* __restrict__ nf, const float* __restrict__ ef,
    const float* __restrict__ We, const float* __restrict__ be,
    const float* __restrict__ Wm, const float* __restrict__ bm,
    const float* __restrict__ ex, const float* __restrict__ den,
    const int* __restrict__ src, const int* __restrict__ dst,
    float* __restrict__ Cv) {
  __shared__ float ein[16][NF + EF];
  __shared__ float he[16][EH];
  __shared__ float alph[16];
  __shared__ int ddst[16];
  int tid = threadIdx.x;
  int ebase = blockIdx.x * 16;
  int er = tid >> 4, seg = tid & 15;
  int e = ebase + er;
  int s = src[e];
  if (seg < 8) {
    const float* p = nf + s * NF + seg * 8;
#pragma unroll
    for (int q = 0; q < 8; ++q) ein[er][seg * 8 + q] = p[q];
  } else {
    const float* p = ef + e * EF + (seg - 8) * 8;
#pragma unroll
    for (int q = 0; q < 8; ++q) ein[er][seg * 8 + q] = p[q];
  }
  if (tid < 16) {
    int ee = ebase + tid;
    int d = dst[ee];
    ddst[tid] = d;
    alph[tid] = ex[ee] / den[d];
  }
  __syncthreads();

  int wave = tid >> 5, lane = tid & 31;
  int m = lane & 15, half = lane >> 4;
  int ncol = wave * 16 + m;
  v8f acc = {};
  for (int k = 0; k < NF + EF; k += 4) {
    int ka = k + half * 2;
    v2f a, b;
    a.x = ein[m][ka];
    a.y = ein[m][ka + 1];
    b.x = We[ka * EH + ncol];
    b.y = We[(ka + 1) * EH + ncol];
    acc = wmma4(a, b, acc);
  }
  float ebias = be[ncol];
#pragma unroll
  for (int i = 0; i < 8; ++i) he[i + half * 8][ncol] = lrelu(acc[i] + ebias);
  __syncthreads();

  // m = h_e @ W_msg + b_msg ; scatter alpha*m into C_v[dst]
  v8f mc = {};
  for (int k = 0; k < EH; k += 4) {
    int ka = k + half * 2;
    v2f a, b;
    a.x = he[m][ka];
    a.y = he[m][ka + 1];
    b.x = Wm[ka * CS + ncol];
    b.y = Wm[(ka + 1) * CS + ncol];
    mc = wmma4(a, b, mc);
  }
  float mbias = bm[ncol];
#pragma unroll
  for (int i = 0; i < 8; ++i) {
    int row = i + half * 8;
    float v = (mc[i] + mbias) * alph[row];
    atomicAdd(&Cv[ddst[row] * CS + ncol], v);
  }
}

// ---------------------------------------------------------------------------
// (6) GRUCell(input=elu(C_v), hidden=h_v) + ReLU  -> d_out[0 : 10000*128]
// 16 nodes per workgroup; 48 WMMA column tiles (24 gi + 24 gh) over 8 waves.
// ---------------------------------------------------------------------------
__global__ __launch_bounds__(256) void gru_k(
    const float* __restrict__ Cv, const float* __restrict__ hv,
    const float* __restrict__ Wih, const float* __restrict__ Whh,
    const float* __restrict__ bih, const float* __restrict__ bhh,
    float* __restrict__ out) {
  __shared__ float ac[16][CS];       // elu(C_v) tile
  __shared__ float ah[16][NH];       // h_v tile
  __shared__ float gi[16][3 * NH];
  __shared__ float gh[16][3 * NH];
  int tid = threadIdx.x;
  int base = blockIdx.x * 16;
  int nr = tid >> 4, seg = tid & 15;
  {
    const float* pc = Cv + (base + nr) * CS + seg * 8;
    const float* ph = hv + (base + nr) * NH + seg * 8;
#pragma unroll
    for (int q = 0; q < 8; ++q) {
      float v = pc[q];
      ac[nr][seg * 8 + q] = v > 0.0f ? v : expm1f(v);  // elu
      ah[nr][seg * 8 + q] = ph[q];
    }
  }
  __syncthreads();

  int wave = tid >> 5, lane = tid & 31;
  int m = lane & 15, half = lane >> 4;
  for (int it = 0; it < 6; ++it) {
    int tile = wave + 8 * it;  // 0..47 (uniform per wave)
    bool isGi = tile < 24;
    int j0 = (isGi ? tile : tile - 24) * 16;
    const float* W = isGi ? Wih : Whh;  // [384][128]; B[k][j] = W[j*128+k]
    v8f acc = {};
    for (int k = 0; k < NH; k += 4) {
      int ka = k + half * 2;
      v2f a, b;
      a.x = isGi ? ac[m][ka] : ah[m][ka];
      a.y = isGi ? ac[m][ka + 1] : ah[m][ka + 1];
      b.x = W[(j0 + m) * NH + ka];
      b.y = W[(j0 + m) * NH + ka + 1];
      acc = wmma4(a, b, acc);
    }
    float bias = isGi ? bih[j0 + m] : bhh[j0 + m];
#pragma unroll
    for (int i = 0; i < 8; ++i) {
      if (isGi) gi[i + half * 8][j0 + m] = acc[i] + bias;
      else      gh[i + half * 8][j0 + m] = acc[i] + bias;
    }
  }
  __syncthreads();

#pragma unroll
  for (int q = 0; q < 8; ++q) {
    int ch = seg * 8 + q;
    float r = 1.0f / (1.0f + __expf(-(gi[nr][ch] + gh[nr][ch])));
    float z = 1.0f / (1.0f + __expf(-(gi[nr][NH + ch] + gh[nr][NH + ch])));
    float nn = tanhf(gi[nr][2 * NH + ch] + r * gh[nr][2 * NH + ch]);
    float h = (1.0f - z) * nn + z * ah[nr][ch];
    out[(base + nr) * NH + ch] = h > 0.0f ? h : 0.0f;
  }
}

// ---------------------------------------------------------------------------
extern "C" void kernel_launch(void* const* d_in, const int* in_sizes, int n_in,
                              void* d_out, int out_size, void* d_ws, size_t ws_size,
                              hipStream_t stream) {
  const float* node_feats = (const float*)d_in[0];
  const float* edge_feats = (const float*)d_in[1];
  const float* W_node  = (const float*)d_in[2];
  const float* b_node  = (const float*)d_in[3];
  const float* W_edge  = (const float*)d_in[4];
  const float* b_edge  = (const float*)d_in[5];
  const float* W_logit = (const float*)d_in[6];
  const float* b_logit = (const float*)d_in[7];
  const float* W_msg   = (const float*)d_in[8];
  const float* b_msg   = (const float*)d_in[9];
  const float* W_ih    = (const float*)d_in[10];
  const float* W_hh    = (const float*)d_in[11];
  const float* b_ih    = (const float*)d_in[12];
  const float* b_hh    = (const float*)d_in[13];
  const int*   src     = (const int*)d_in[14];
  const int*   dst     = (const int*)d_in[15];

  // workspace layout (floats): ~13 MB total
  float* ws      = (float*)d_ws;
  float* h_v     = ws;                       // 1,280,000
  float* logit   = ws + 1280000;             //   640,000 (later holds ex)
  float* den     = ws + 1920000;             //    10,000
  unsigned* lmax = (unsigned*)(ws + 1930000);//    10,000
  float* Cv      = ws + 1940000;             // 1,280,000

  float* out_h = (float*)d_out;
  float* out_e = out_h + (size_t)N_NODES * NH;

  init_k<<<(N_NODES * CS + 255) / 256, 256, 0, stream>>>(Cv, den, lmax);
  node_embed_k<<<N_NODES / 16, 256, 0, stream>>>(node_feats, W_node, b_node, h_v);
  edge_logit_k<<<N_EDGES / 16, 256, 0, stream>>>(node_feats, edge_feats, W_edge,
                                                 b_edge, W_logit, b_logit, h_v,
                                                 src, logit);
  seg_max_k<<<N_EDGES / 256, 256, 0, stream>>>(logit, dst, lmax);
  seg_exp_k<<<N_EDGES / 256, 256, 0, stream>>>(logit, dst, lmax, den);
  edge_scatter_k<<<N_EDGES / 16, 256, 0, stream>>>(node_feats, edge_feats, W_edge,
                                                   b_edge, W_msg, b_msg, logit,
                                                   den, src, dst, Cv);
  gru_k<<<N_NODES / 16, 256, 0, stream>>>(Cv, h_v, W_ih, W_hh, b_ih, b_hh, out_h);
  hipMemcpyAsync(out_e, edge_feats, (size_t)N_EDGES * EF * sizeof(float),
                 hipMemcpyDeviceToDevice, stream);
}